// KeyQueryAttention_89026082111905
// MI455X (gfx1250) — compile-verified
//
#include <hip/hip_runtime.h>
#include <hip/hip_bf16.h>

#define B_  64
#define T_  8192
#define D_  128
#define L_  64
#define NC  64                      // chunks per batch
#define RPC (T_ / NC)               // 128 rows per chunk
#define PSTRIDE 132                 // m, s, pad, pad, acc[128]  (528B = 33*16, keeps acc 16B-aligned)
#define PART_FLOATS ((size_t)B_ * NC * PSTRIDE)

typedef __attribute__((ext_vector_type(2))) float v2f;
typedef __attribute__((ext_vector_type(4))) float v4f;
typedef __attribute__((ext_vector_type(8))) float v8f;

static __device__ __forceinline__ float neg_big() { return -3.402823466e38f; }

// ---------------------------------------------------------------------------
// Pack kw/qw into WMMA-B-fragment order so the hot loop issues one b128 load
// per (k-step, col-tile) instead of four b32 loads:
//   wpack[(n*64 + p)*4 + {0,1,2,3}] = { kw[2p][n], kw[2p+1][n], qw[2p][n], qw[2p+1][n] }
// n = 0..63 (output col), p = 0..63 (k-pair index)
// ---------------------------------------------------------------------------
__global__ __launch_bounds__(256)
void kqa_pack(const float* __restrict__ kw, const float* __restrict__ qw,
              float* __restrict__ wpack)
{
    const int idx = blockIdx.x * blockDim.x + threadIdx.x;   // 0..4095
    const int n = idx >> 6;
    const int p = idx & 63;
    const int k = 2 * p;
    v4f w;
    w.x = kw[(size_t)k * L_ + n];
    w.y = kw[(size_t)(k + 1) * L_ + n];
    w.z = qw[(size_t)k * L_ + n];
    w.w = qw[(size_t)(k + 1) * L_ + n];
    *(v4f*)(wpack + (size_t)idx * 4) = w;
}

// ---------------------------------------------------------------------------
// Pass 1: one wave per (batch, chunk). Fused keys/queries GEMM via
// V_WMMA_F32_16X16X4_F32 + logits + online masked softmax + weighted pooling.
// ---------------------------------------------------------------------------
__global__ __launch_bounds__(32)
void kqa_pass1(const float* __restrict__ seq,
               const long long* __restrict__ lengths,
               const float* __restrict__ wpack,
               float* __restrict__ part)
{
    const int b    = blockIdx.x / NC;
    const int c    = blockIdx.x % NC;
    const int lane = threadIdx.x;      // 0..31
    const int lo   = lane & 15;
    const int hi   = lane >> 4;        // 0 = low half, 1 = high half

    const long long len64 = lengths[b];
    const int base  = c * RPC;
    long long lim64 = (long long)(base + RPC);
    if (len64 < lim64) lim64 = len64;
    const int limit = (int)lim64;
    const int len   = (len64 > (long long)T_) ? T_ : (int)len64;

    const float* __restrict__ seqb = seq + (size_t)b * T_ * D_;

    float m = neg_big();
    float s = 0.0f;
    v4f acc = (v4f){0.f, 0.f, 0.f, 0.f};           // d = 4*lane + j

    for (int t0 = base; t0 < limit; t0 += 16) {
        v8f ck[4], cq[4];
        #pragma unroll
        for (int ct = 0; ct < 4; ++ct) {
            ck[ct] = (v8f){0.f,0.f,0.f,0.f,0.f,0.f,0.f,0.f};
            cq[ct] = (v8f){0.f,0.f,0.f,0.f,0.f,0.f,0.f,0.f};
        }

        // A fragment: row = t0 + lo, K pair at offset 2*hi (ISA 16x4 layout)
        const float* __restrict__ arow = seqb + (size_t)(t0 + lo) * D_ + 2 * hi;

        #pragma unroll 2
        for (int k0 = 0; k0 < D_; k0 += 4) {
            const v2f a = *(const v2f*)(arow + k0);           // b64
            const int pidx = (k0 >> 1) + hi;                  // k-pair index
            #pragma unroll
            for (int ct = 0; ct < 4; ++ct) {
                const int n = ct * 16 + lo;
                const v4f w4 = *(const v4f*)(wpack + ((size_t)n * 64 + pidx) * 4); // b128
                v2f bk, bq;
                bk.x = w4.x; bk.y = w4.y;
                bq.x = w4.z; bq.y = w4.w;
                ck[ct] = __builtin_amdgcn_wmma_f32_16x16x4_f32(
                             false, a, false, bk, (short)0, ck[ct], false, false);
                cq[ct] = __builtin_amdgcn_wmma_f32_16x16x4_f32(
                             false, a, false, bq, (short)0, cq[ct], false, false);
            }
        }

        // logits: row M = v + 8*hi lives in VGPR v across this half's 16 lanes
        float lg[8];
        #pragma unroll
        for (int v = 0; v < 8; ++v) {
            float p = 0.f;
            #pragma unroll
            for (int ct = 0; ct < 4; ++ct) p += ck[ct][v] * cq[ct][v];
            p += __shfl_xor(p, 1, 32);
            p += __shfl_xor(p, 2, 32);
            p += __shfl_xor(p, 4, 32);
            p += __shfl_xor(p, 8, 32);
            const int row = t0 + v + 8 * hi;
            lg[v] = (row < len) ? p : neg_big();
        }

        // online softmax update (wave-uniform)
        float tmax = lg[0];
        #pragma unroll
        for (int v = 1; v < 8; ++v) tmax = fmaxf(tmax, lg[v]);
        tmax = fmaxf(tmax, __shfl_xor(tmax, 16, 32));

        const float nm    = fmaxf(m, tmax);
        const float scale = __expf(m - nm);
        acc.x *= scale; acc.y *= scale; acc.z *= scale; acc.w *= scale;

        float w[8];
        float tsum = 0.f;
        #pragma unroll
        for (int v = 0; v < 8; ++v) { w[v] = __expf(lg[v] - nm); tsum += w[v]; }
        tsum += __shfl_xor(tsum, 16, 32);
        s = s * scale + tsum;
        m = nm;

        // weighted accumulation: one b128 per row (cache-hot reload of tile)
        const float* __restrict__ rowp = seqb + (size_t)t0 * D_ + 4 * lane;
        #pragma unroll
        for (int i = 0; i < 16; ++i) {
            const float wi = __shfl(w[i & 7], (i >> 3) << 4, 32);
            const v4f r = *(const v4f*)(rowp + (size_t)i * D_);
            acc.x = fmaf(wi, r.x, acc.x);
            acc.y = fmaf(wi, r.y, acc.y);
            acc.z = fmaf(wi, r.z, acc.z);
            acc.w = fmaf(wi, r.w, acc.w);
        }
    }

    // emit partial: [m, s, pad, pad, acc[128]]
    float* __restrict__ p = part + ((size_t)b * NC + c) * PSTRIDE;
    if (lane == 0) { p[0] = m; p[1] = s; }
    *(v4f*)(p + 4 + 4 * lane) = acc;               // 16B-aligned b128 store
}

// ---------------------------------------------------------------------------
// Pass 2: merge NC partials per batch; out[b,d] = A/S + bias[d]
// ---------------------------------------------------------------------------
__global__ __launch_bounds__(128)
void kqa_pass2(const float* __restrict__ part,
               const float* __restrict__ bias,
               float* __restrict__ out)
{
    const int b = blockIdx.x;
    const int d = threadIdx.x;          // 0..127
    const float* __restrict__ pb = part + (size_t)b * NC * PSTRIDE;

    float M = neg_big();
    for (int c = 0; c < NC; ++c) M = fmaxf(M, pb[(size_t)c * PSTRIDE]);

    float S = 0.f, A = 0.f;
    for (int c = 0; c < NC; ++c) {
        const float* p = pb + (size_t)c * PSTRIDE;
        const float e = __expf(p[0] - M);
        S += p[1] * e;
        A = fmaf(p[4 + d], e, A);
    }
    out[(size_t)b * D_ + d] = A / S + bias[d];
}

// ---------------------------------------------------------------------------
extern "C" void kernel_launch(void* const* d_in, const int* in_sizes, int n_in,
                              void* d_out, int out_size, void* d_ws, size_t ws_size,
                              hipStream_t stream)
{
    (void)in_sizes; (void)n_in; (void)out_size; (void)ws_size;
    const float*     seq     = (const float*)d_in[0];
    const long long* lengths = (const long long*)d_in[1];   // jnp.int64
    const float*     key_w   = (const float*)d_in[2];
    const float*     query_w = (const float*)d_in[3];
    const float*     bias    = (const float*)d_in[4];
    float* out   = (float*)d_out;
    float* part  = (float*)d_ws;                    // 64*64*132 floats ≈ 2.16 MB
    float* wpack = (float*)d_ws + PART_FLOATS;      // + 64 KB packed weights

    kqa_pack <<<16, 256, 0, stream>>>(key_w, query_w, wpack);
    kqa_pass1<<<B_ * NC, 32, 0, stream>>>(seq, lengths, wpack, part);
    kqa_pass2<<<B_, 128, 0, stream>>>(part, bias, out);
}